// MultiHeadAttention_17867063951852
// MI455X (gfx1250) — compile-verified
//
#include <hip/hip_runtime.h>
#include <hip/hip_bf16.h>

// ---------------------------------------------------------------------------
// MHA forward for gfx1250 (MI455X), fp32 end-to-end via V_WMMA_F32_16X16X4_F32.
// seq=4096, d_in=d_out=512, 8 heads x 64. One wave (32 threads) per block so
// EXEC is all-ones at every WMMA (ISA requirement).
//
// fp32 WMMA register layouts used (CDNA5 ISA 7.12.2):
//   A 16x4 :  lane L(0-15)=row M=L ; v.x = K = 2*(lane>>4), v.y = K+1
//   B 4x16 :  col N = lane&15 ; v.x = row K = (lane>>4), v.y = row K+2
//   C 16x16:  VGPR r -> M = r + 8*(lane>>4) ; N = lane&15
//
// Attention K/V tiles are staged through LDS with the CDNA5 async path
// (GLOBAL_LOAD_ASYNC_TO_LDS_B128 + s_wait_asynccnt), double-buffered so the
// next 16-key tile streams in while WMMAs consume the current one.
// ---------------------------------------------------------------------------

typedef __attribute__((ext_vector_type(2))) float v2f;
typedef __attribute__((ext_vector_type(8))) float v8f;

#define SEQ  4096
#define DIN  512
#define DOUT 512
#define NH   8
#define HD   64

static __device__ __forceinline__ v8f wmma_f32(v2f a, v2f b, v8f c) {
  // (neg_a, A, neg_b, B, c_mod, C, reuse_a, reuse_b)
  return __builtin_amdgcn_wmma_f32_16x16x4_f32(false, a, false, b, (short)0, c,
                                               false, false);
}

// Async-copy one contiguous 4 KB tile (16 rows x 64 f32) global -> LDS.
// Each of the 32 lanes moves 8 x 16B. Tracked on ASYNCcnt (16 ops/tile-pair).
// LDS byte offset = low 32 bits of the generic pointer (ISA 10.2 aperture map).
static __device__ __forceinline__ void async_tile_4k(const float* __restrict__ g,
                                                     const float* l, int lane) {
  unsigned long long ga = (unsigned long long)(uintptr_t)g + (unsigned)(lane * 16);
  unsigned la = (unsigned)(uintptr_t)l + (unsigned)(lane * 16);
#pragma unroll
  for (int i = 0; i < 8; ++i) {
    asm volatile("global_load_async_to_lds_b128 %0, %1, off"
                 :
                 : "v"(la + (unsigned)(i * 512)),
                   "v"(ga + (unsigned long long)(i * 512))
                 : "memory");
  }
}

// ---------------------------------------------------------------------------
// Kernel 1: fused QKV projection.  C = X @ W^T + b, one of {q,k,v} per grid.z.
// 64x64 output tile per wave; output scattered into [head][seq][64] layout.
// ---------------------------------------------------------------------------
__global__ __launch_bounds__(32) void qkv_gemm(
    const float* __restrict__ X,
    const float* __restrict__ Wq, const float* __restrict__ bq,
    const float* __restrict__ Wk, const float* __restrict__ bk,
    const float* __restrict__ Wv, const float* __restrict__ bv,
    float* __restrict__ Qo, float* __restrict__ Ko, float* __restrict__ Vo) {
  const int lane = threadIdx.x;
  const int row  = lane & 15;
  const int hi   = lane >> 4;
  const int mBase = blockIdx.y * 64;
  const int nBase = blockIdx.x * 64;

  const float* W; const float* bias; float* dst;
  if (blockIdx.z == 0)      { W = Wq; bias = bq; dst = Qo; }
  else if (blockIdx.z == 1) { W = Wk; bias = bk; dst = Ko; }
  else                      { W = Wv; bias = bv; dst = Vo; }

  const v8f vzero = {0.f, 0.f, 0.f, 0.f, 0.f, 0.f, 0.f, 0.f};
  v8f acc[4][4];
#pragma unroll
  for (int mt = 0; mt < 4; ++mt)
#pragma unroll
    for (int nt = 0; nt < 4; ++nt) acc[mt][nt] = vzero;

  for (int k0 = 0; k0 < DIN; k0 += 4) {
    v2f a[4], bf[4];
#pragma unroll
    for (int mt = 0; mt < 4; ++mt) {
      const float* xp = X + (size_t)(mBase + mt * 16 + row) * DIN + k0 + 2 * hi;
      a[mt].x = xp[0];
      a[mt].y = xp[1];
    }
#pragma unroll
    for (int nt = 0; nt < 4; ++nt) {
      // B[k][n] = W[nBase+nt*16+n][k]   (C = X * W^T)
      const float* wp = W + (size_t)(nBase + nt * 16 + row) * DIN + k0 + hi;
      bf[nt].x = wp[0];   // K = k0 + hi
      bf[nt].y = wp[2];   // K = k0 + 2 + hi
    }
#pragma unroll
    for (int mt = 0; mt < 4; ++mt)
#pragma unroll
      for (int nt = 0; nt < 4; ++nt)
        acc[mt][nt] = wmma_f32(a[mt], bf[nt], acc[mt][nt]);
  }

  // Epilogue: bias + scatter into [head][seq][hd]
#pragma unroll
  for (int nt = 0; nt < 4; ++nt) {
    const int n  = nBase + nt * 16 + row;
    const int h  = n >> 6;
    const int d  = n & 63;
    const float bn = bias[n];
#pragma unroll
    for (int mt = 0; mt < 4; ++mt) {
#pragma unroll
      for (int r = 0; r < 8; ++r) {
        const int m = mBase + mt * 16 + r + 8 * hi;
        dst[((size_t)h * SEQ + m) * HD + d] = acc[mt][nt][r] + bn;
      }
    }
  }
}

// ---------------------------------------------------------------------------
// Kernel 2: causal flash attention.  One wave per (16-query block, head).
// K/V tiles double-buffered through LDS via async-to-LDS b128 transfers.
// ---------------------------------------------------------------------------
__global__ __launch_bounds__(32) void attn_kernel(
    const float* __restrict__ Q, const float* __restrict__ K,
    const float* __restrict__ V, float* __restrict__ O) {
  __shared__ float kbuf[2][16 * HD];  // 4 KB per buffer
  __shared__ float vbuf[2][16 * HD];  // 4 KB per buffer
  __shared__ float plds[16 * 17];     // P tile staging (C-layout -> A-layout)

  const int lane  = threadIdx.x;
  const int row   = lane & 15;
  const int hi    = lane >> 4;
  const int qt    = blockIdx.x;
  const int h     = blockIdx.y;
  const int qBase = qt * 16;

  const float* Qh = Q + (size_t)h * SEQ * HD;
  const float* Kh = K + (size_t)h * SEQ * HD;
  const float* Vh = V + (size_t)h * SEQ * HD;

  // Preload Q A-fragments with 1/sqrt(64) folded in.
  v2f qa[16];
#pragma unroll
  for (int kk = 0; kk < 16; ++kk) {
    const float* qp = Qh + (size_t)(qBase + row) * HD + kk * 4 + 2 * hi;
    qa[kk].x = qp[0] * 0.125f;
    qa[kk].y = qp[1] * 0.125f;
  }

  const v8f vzero = {0.f, 0.f, 0.f, 0.f, 0.f, 0.f, 0.f, 0.f};
  v8f oacc[4];
#pragma unroll
  for (int ct = 0; ct < 4; ++ct) oacc[ct] = vzero;
  float m_i[8], l_i[8];
#pragma unroll
  for (int r = 0; r < 8; ++r) { m_i[r] = -__builtin_inff(); l_i[r] = 0.f; }

  // Prologue: stream first K/V tile into LDS buffer 0.
  async_tile_4k(Kh, kbuf[0], lane);
  async_tile_4k(Vh, vbuf[0], lane);

  for (int jt = 0; jt <= qt; ++jt) {
    const int buf = jt & 1;
    const float* kt = kbuf[buf];
    const float* vt = vbuf[buf];

    // Prefetch next tile into the other buffer, then wait for the current
    // tile's 16 async ops (in-order completion => ASYNCcnt<=16 suffices).
    // WAR on the overwritten buffer is safe: last iteration's ds_load->wmma
    // register deps forced s_wait_dscnt before this point.
    if (jt < qt) {
      async_tile_4k(Kh + (size_t)(jt + 1) * 16 * HD, kbuf[buf ^ 1], lane);
      async_tile_4k(Vh + (size_t)(jt + 1) * 16 * HD, vbuf[buf ^ 1], lane);
      asm volatile("s_wait_asynccnt 0x10" ::: "memory");
    } else {
      asm volatile("s_wait_asynccnt 0x0" ::: "memory");
    }

    // S = (Q/sqrt(hd)) . K^T  : B[d][n] = Ktile[n][d]   (from LDS)
    v8f s = vzero;
#pragma unroll
    for (int kk = 0; kk < 16; ++kk) {
      v2f kb;
      kb.x = kt[row * HD + kk * 4 + hi];       // d = kk*4 + hi
      kb.y = kt[row * HD + kk * 4 + 2 + hi];   // d = kk*4 + 2 + hi
      s = wmma_f32(qa[kk], kb, s);
    }

    const bool diag = (jt == qt);

    // Online softmax per owned row (C-layout: row = r + 8*hi, col = lane&15).
#pragma unroll
    for (int r = 0; r < 8; ++r) {
      const int mrow = r + 8 * hi;
      float sv = s[r];
      if (diag && (row > mrow)) sv = -__builtin_inff();   // causal mask

      float mx = sv;                      // row-max across the 16-lane half
      mx = fmaxf(mx, __shfl_xor(mx, 1, 32));
      mx = fmaxf(mx, __shfl_xor(mx, 2, 32));
      mx = fmaxf(mx, __shfl_xor(mx, 4, 32));
      mx = fmaxf(mx, __shfl_xor(mx, 8, 32));

      const float mnew = fmaxf(m_i[r], mx);
      const float p    = __expf(sv - mnew);
      float rs = p;                       // row-sum across the half
      rs += __shfl_xor(rs, 1, 32);
      rs += __shfl_xor(rs, 2, 32);
      rs += __shfl_xor(rs, 4, 32);
      rs += __shfl_xor(rs, 8, 32);

      const float alpha = __expf(m_i[r] - mnew);
      l_i[r] = l_i[r] * alpha + rs;
      m_i[r] = mnew;
#pragma unroll
      for (int ct = 0; ct < 4; ++ct) oacc[ct][r] *= alpha;

      plds[mrow * 17 + row] = p;          // stage P for layout conversion
    }
    __syncthreads();

    // O += P . V  : A = P (16x16 from LDS in A-layout), B[k][n] = Vtile[k][n]
#pragma unroll
    for (int kk = 0; kk < 4; ++kk) {
      v2f pa;
      pa.x = plds[row * 17 + kk * 4 + 2 * hi];
      pa.y = plds[row * 17 + kk * 4 + 2 * hi + 1];
#pragma unroll
      for (int ct = 0; ct < 4; ++ct) {
        v2f vb;
        vb.x = vt[(kk * 4 + hi) * HD + ct * 16 + row];       // key kk*4 + hi
        vb.y = vt[(kk * 4 + 2 + hi) * HD + ct * 16 + row];   // key kk*4+2+hi
        oacc[ct] = wmma_f32(pa, vb, oacc[ct]);
      }
    }
    __syncthreads();
  }

  // Normalize and write [head][seq][hd]
#pragma unroll
  for (int r = 0; r < 8; ++r) {
    const float inv  = 1.0f / l_i[r];
    const int   mrow = r + 8 * hi;
#pragma unroll
    for (int ct = 0; ct < 4; ++ct)
      O[((size_t)h * SEQ + qBase + mrow) * HD + ct * 16 + row] =
          oacc[ct][r] * inv;
  }
}

// ---------------------------------------------------------------------------
// Kernel 3: output projection.  out = concat(O) @ Wo^T + bo, row-major out.
// A element (i,k) lives at ows[(k>>6)*SEQ*HD + i*HD + (k&63)].
// ---------------------------------------------------------------------------
__global__ __launch_bounds__(32) void out_gemm(
    const float* __restrict__ A, const float* __restrict__ Wo,
    const float* __restrict__ bo, float* __restrict__ out) {
  const int lane = threadIdx.x;
  const int row  = lane & 15;
  const int hi   = lane >> 4;
  const int mBase = blockIdx.y * 64;
  const int nBase = blockIdx.x * 64;

  const v8f vzero = {0.f, 0.f, 0.f, 0.f, 0.f, 0.f, 0.f, 0.f};
  v8f acc[4][4];
#pragma unroll
  for (int mt = 0; mt < 4; ++mt)
#pragma unroll
    for (int nt = 0; nt < 4; ++nt) acc[mt][nt] = vzero;

  for (int k0 = 0; k0 < DOUT; k0 += 4) {
    const int hsrc = k0 >> 6;           // head index (constant for the 4-group)
    const int dbase = (k0 & 63) + 2 * hi;
    v2f a[4], bf[4];
#pragma unroll
    for (int mt = 0; mt < 4; ++mt) {
      const float* ap =
          A + ((size_t)hsrc * SEQ + (mBase + mt * 16 + row)) * HD + dbase;
      a[mt].x = ap[0];
      a[mt].y = ap[1];
    }
#pragma unroll
    for (int nt = 0; nt < 4; ++nt) {
      const float* wp = Wo + (size_t)(nBase + nt * 16 + row) * DOUT + k0 + hi;
      bf[nt].x = wp[0];
      bf[nt].y = wp[2];
    }
#pragma unroll
    for (int mt = 0; mt < 4; ++mt)
#pragma unroll
      for (int nt = 0; nt < 4; ++nt)
        acc[mt][nt] = wmma_f32(a[mt], bf[nt], acc[mt][nt]);
  }

#pragma unroll
  for (int nt = 0; nt < 4; ++nt) {
    const int n = nBase + nt * 16 + row;
    const float bn = bo[n];
#pragma unroll
    for (int mt = 0; mt < 4; ++mt) {
#pragma unroll
      for (int r = 0; r < 8; ++r) {
        const int m = mBase + mt * 16 + r + 8 * hi;
        out[(size_t)m * DOUT + n] = acc[mt][nt][r] + bn;
      }
    }
  }
}

// ---------------------------------------------------------------------------
extern "C" void kernel_launch(void* const* d_in, const int* in_sizes, int n_in,
                              void* d_out, int out_size, void* d_ws,
                              size_t ws_size, hipStream_t stream) {
  const float* X  = (const float*)d_in[0];
  const float* Wq = (const float*)d_in[1];
  const float* bq = (const float*)d_in[2];
  const float* Wk = (const float*)d_in[3];
  const float* bk = (const float*)d_in[4];
  const float* Wv = (const float*)d_in[5];
  const float* bv = (const float*)d_in[6];
  const float* Wo = (const float*)d_in[7];
  const float* bo = (const float*)d_in[8];
  float* out = (float*)d_out;

  // Workspace partition: Q,K,V,O each SEQ*DOUT floats (8 MB) -> 32 MB total.
  float* qw = (float*)d_ws;
  float* kw = qw + (size_t)SEQ * DOUT;
  float* vw = kw + (size_t)SEQ * DOUT;
  float* ow = vw + (size_t)SEQ * DOUT;

  // 1) QKV projections: grid (N/64, M/64, 3), one 32-lane wave per block.
  dim3 g1(DOUT / 64, SEQ / 64, 3);
  qkv_gemm<<<g1, 32, 0, stream>>>(X, Wq, bq, Wk, bk, Wv, bv, qw, kw, vw);

  // 2) Causal flash attention: grid (SEQ/16 query blocks, 8 heads).
  dim3 g2(SEQ / 16, NH, 1);
  attn_kernel<<<g2, 32, 0, stream>>>(qw, kw, vw, ow);

  // 3) Output projection.
  dim3 g3(DOUT / 64, SEQ / 64, 1);
  out_gemm<<<g3, 32, 0, stream>>>(ow, Wo, bo, out);
}